// GraphConv_103079215779
// MI455X (gfx1250) — compile-verified
//
#include <hip/hip_runtime.h>
#include <math.h>

#define N_USERS   100000
#define N_ITEMS   50000
#define N_ENTITIES 150000
#define N_EDGES   2000000
#define NNZ       1000000
#define CH        64
#define TEMP_INV  5.0f        // 1/0.2
#define PDIST_EPS 1e-6f
#define NORM_EPS  1e-12f
#define EG        4           // edges gathered per wave via async-LDS

typedef __attribute__((ext_vector_type(2))) float v2f;
typedef __attribute__((ext_vector_type(8))) float v8f;

// ---------------------------------------------------------------- utilities
__global__ __launch_bounds__(256) void k_zero(float* p, long n) {
  long i = (long)blockIdx.x * blockDim.x + threadIdx.x;
  long stride = (long)gridDim.x * blockDim.x;
  for (; i < n; i += stride) p[i] = 0.0f;
}

// out = [user_emb | entity_emb] as the residual starting point
__global__ __launch_bounds__(256) void k_init_out(const float4* __restrict__ ue,
                                                  const float4* __restrict__ ee,
                                                  float4* __restrict__ out) {
  int i = blockIdx.x * blockDim.x + threadIdx.x;   // 4,000,000 threads
  const int NU4 = N_USERS * CH / 4;
  if (i < NU4) out[i] = ue[i];
  else         out[i] = ee[i - NU4];
}

// ------------------------------------------------ edge gather + scatter-add
// wave handles EG edges: async-gather EG entity rows (256B each) into LDS,
// wait, then scatter-add into summed[] / cnt[] with f32 atomics (L2-resident).
__global__ __launch_bounds__(256) void k_edge_scatter(const float* __restrict__ ent,
                                                      const int* __restrict__ head,
                                                      const int* __restrict__ tail,
                                                      float* __restrict__ summed,
                                                      float* __restrict__ cnt) {
  __shared__ float sh[8 * EG * CH];                 // 8 waves * 4 rows * 64f = 8KB
  const int tid  = threadIdx.x;
  const int lane = tid & 31;
  const int w    = tid >> 5;
  const long group = (long)blockIdx.x * 8 + w;      // N_EDGES/EG groups total
  const long e0 = group * EG;
  const unsigned lds_base = (unsigned)(w * EG * CH * 4);  // bytes (sh at LDS offset 0)

#pragma unroll
  for (int e = 0; e < EG; ++e) {
    const int t = tail[e0 + e];
    unsigned long long ga =
        (unsigned long long)(uintptr_t)(ent + (long)t * CH) + (unsigned)(lane * 8);
    unsigned la = lds_base + (unsigned)(e * CH * 4 + lane * 8);
    // CDNA5 async DMA: 32 lanes x 8B = one 256B embedding row -> LDS
    asm volatile("global_load_async_to_lds_b64 %0, %1, off"
                 :: "v"(la), "v"(ga) : "memory");
  }
  asm volatile("s_wait_asynccnt 0x0" ::: "memory");

#pragma unroll
  for (int e = 0; e < EG; ++e) {
    const int h = head[e0 + e];
    float2 v = *(const float2*)&sh[w * EG * CH + e * CH + lane * 2];
    float* dst = summed + (long)h * CH + lane * 2;
    atomicAdd(dst,     v.x);
    atomicAdd(dst + 1, v.y);
    if (lane == 0) atomicAdd(cnt + h, 1.0f);
  }
}

// entity_agg = summed / max(cnt,1)   (in place, float4 per thread)
__global__ __launch_bounds__(256) void k_entity_div(float4* __restrict__ agg,
                                                    const float* __restrict__ cnt) {
  int i = blockIdx.x * blockDim.x + threadIdx.x;    // N_ENTITIES*16 threads
  int row = i >> 4;
  float s = 1.0f / fmaxf(cnt[row], 1.0f);
  float4 v = agg[i];
  v.x *= s; v.y *= s; v.z *= s; v.w *= s;
  agg[i] = v;
}

// ---------------------------------- weighted scatter into user accumulators
// mode 0: coef = vals[nz]              -> user_mean
// mode 1: coef = e[nz] / denom[user]   -> user_agg
__global__ __launch_bounds__(256) void k_coef_scatter(const int* __restrict__ iu,
                                                      const int* __restrict__ ii,
                                                      const float* __restrict__ vals,
                                                      const float* __restrict__ escore,
                                                      const float* __restrict__ denom,
                                                      const float* __restrict__ agg,
                                                      float* __restrict__ dest,
                                                      int mode) {
  long wid = ((long)blockIdx.x * blockDim.x + threadIdx.x) >> 5;
  int lane = threadIdx.x & 31;
  if (wid >= NNZ) return;
  int u = iu[wid], it = ii[wid];
  float coef = mode ? (escore[wid] / denom[u]) : vals[wid];
  float2 g = *(const float2*)(agg + (long)it * CH + lane * 2);
  float* d = dest + (long)u * CH + lane * 2;
  atomicAdd(d,     coef * g.x);
  atomicAdd(d + 1, coef * g.y);
}

// score[nz] = ||g - user_mean + eps|| / T ; rowmax = segment max (bits of f32>=0)
__global__ __launch_bounds__(256) void k_score(const int* __restrict__ iu,
                                               const int* __restrict__ ii,
                                               const float* __restrict__ agg,
                                               const float* __restrict__ umean,
                                               float* __restrict__ score,
                                               unsigned* __restrict__ rowmax) {
  long wid = ((long)blockIdx.x * blockDim.x + threadIdx.x) >> 5;
  int lane = threadIdx.x & 31;
  if (wid >= NNZ) return;
  int u = iu[wid], it = ii[wid];
  float2 g = *(const float2*)(agg + (long)it * CH + lane * 2);
  float2 m = *(const float2*)(umean + (long)u * CH + lane * 2);
  float d0 = g.x - m.x + PDIST_EPS;
  float d1 = g.y - m.y + PDIST_EPS;
  float p = d0 * d0 + d1 * d1;
#pragma unroll
  for (int off = 16; off > 0; off >>= 1) p += __shfl_xor(p, off, 32);
  if (lane == 0) {
    float s = sqrtf(p) * TEMP_INV;                  // s >= 0, so uint-max works
    score[wid] = s;
    atomicMax(rowmax + u, __float_as_uint(s));
  }
}

// e = exp(score - rowmax[u]); denom[u] += e  (score overwritten with e)
__global__ __launch_bounds__(256) void k_exp_denom(const int* __restrict__ iu,
                                                   float* __restrict__ score,
                                                   const unsigned* __restrict__ rowmax,
                                                   float* __restrict__ denom) {
  int i = blockIdx.x * blockDim.x + threadIdx.x;
  if (i >= NNZ) return;
  int u = iu[i];
  float e = expf(score[i] - __uint_as_float(rowmax[u]));
  score[i] = e;
  atomicAdd(denom + u, e);
}

// ------------------------- L2 normalize 16 rows/wave via WMMA, res += norm
// Sum-of-squares: D = (X.^2) x Ones using v_wmma_f32_16x16x4_f32 accumulated
// over 16 K-chunks. B all-ones makes every D column equal to the row sumsq.
__global__ __launch_bounds__(256) void k_norm_acc(const float* __restrict__ src,
                                                  float* __restrict__ dst,
                                                  float* __restrict__ res,
                                                  int nrows) {
  long wid = ((long)blockIdx.x * blockDim.x + threadIdx.x) >> 5;
  int lane = threadIdx.x & 31;
  long r0 = wid * 16;
  if (r0 >= nrows) return;                          // wave-uniform: EXEC stays all-1
  int m  = lane & 15;
  int hi = lane >> 4;

  float sums[8];
#if __has_builtin(__builtin_amdgcn_wmma_f32_16x16x4_f32)
  // A layout (32-bit 16x4): lanes 0-15 -> M=lane, K=0,1 ; lanes 16-31 -> K=2,3
  const float* row = src + (r0 + m) * CH + hi * 2;
  v8f acc = {};
  v2f ones; ones.x = 1.0f; ones.y = 1.0f;
#pragma unroll
  for (int k = 0; k < 16; ++k) {
    float2 x = *(const float2*)(row + 4 * k);
    v2f a; a.x = x.x * x.x; a.y = x.y * x.y;
    acc = __builtin_amdgcn_wmma_f32_16x16x4_f32(false, a, false, ones,
                                                (short)0, acc, false, false);
  }
  // D layout: VGPR r -> row r (lanes 0-15) / row r+8 (lanes 16-31), cols equal
#pragma unroll
  for (int r = 0; r < 8; ++r) sums[r] = acc[r];
#else
#pragma unroll
  for (int r = 0; r < 8; ++r) {
    long rowi = r0 + hi * 8 + r;
    float s = 0.0f;
    for (int c = 0; c < CH; ++c) { float x = src[rowi * CH + c]; s += x * x; }
    sums[r] = s;
  }
#endif

#pragma unroll
  for (int r = 0; r < 8; ++r) {
    long rowi = r0 + hi * 8 + r;                    // lanes<16: rows r0..+7, else +8..+15
    float scale = 1.0f / fmaxf(sqrtf(sums[r]), NORM_EPS);
    float4 v = *((const float4*)(src + rowi * CH) + m);   // lane m -> channels 4m..4m+3
    v.x *= scale; v.y *= scale; v.z *= scale; v.w *= scale;
    if (dst) *((float4*)(dst + rowi * CH) + m) = v;
    float4* rp = (float4*)(res + rowi * CH) + m;
    float4 o = *rp;
    o.x += v.x; o.y += v.y; o.z += v.z; o.w += v.w;
    *rp = o;
  }
}

// ---------------------------------------------------------------- launcher
extern "C" void kernel_launch(void* const* d_in, const int* in_sizes, int n_in,
                              void* d_out, int out_size, void* d_ws, size_t ws_size,
                              hipStream_t stream) {
  const float* user_emb   = (const float*)d_in[0];
  const float* entity_emb = (const float*)d_in[1];
  const int*   edge_index = (const int*)d_in[3];
  const int*   head       = edge_index;             // edge_index[0]
  const int*   tail       = edge_index + N_EDGES;   // edge_index[1]
  const int*   iu         = (const int*)d_in[5];
  const int*   ii         = (const int*)d_in[6];
  const float* vals       = (const float*)d_in[7];

  float* out      = (float*)d_out;
  float* user_res = out;                            // 100000*64
  float* ent_res  = out + (long)N_USERS * CH;       // 150000*64

  // workspace layout (floats)
  float* ws     = (float*)d_ws;
  float* summed = ws;                               // 9,600,000 (becomes entity_agg)
  float* cnt    = summed + (long)N_ENTITIES * CH;   //   150,000
  float* umean  = cnt + N_ENTITIES;                 // 6,400,000
  float* uagg   = umean + (long)N_USERS * CH;       // 6,400,000
  float* denom  = uagg + (long)N_USERS * CH;        //   100,000
  float* rowmax = denom + N_USERS;                  //   100,000
  float* score  = rowmax + N_USERS;                 // 1,000,000
  float* entbuf = score + NNZ;                      // 9,600,000 (normalized ent, next hop)
  const long zero_n = (long)N_ENTITIES * CH + N_ENTITIES +
                      2L * N_USERS * CH + 2L * N_USERS;   // 22,750,000

  k_init_out<<<15625, 256, 0, stream>>>((const float4*)user_emb,
                                        (const float4*)entity_emb, (float4*)out);

  const float* ent_in = entity_emb;
  for (int hop = 0; hop < 2; ++hop) {
    k_zero<<<8192, 256, 0, stream>>>(ws, zero_n);
    k_edge_scatter<<<N_EDGES / (EG * 8), 256, 0, stream>>>(ent_in, head, tail, summed, cnt);
    k_entity_div<<<N_ENTITIES * 16 / 256, 256, 0, stream>>>((float4*)summed, cnt);
    // normalized entities -> entbuf, residual accumulate (reads summed only)
    k_norm_acc<<<(N_ENTITIES / 16 + 7) / 8, 256, 0, stream>>>(summed, entbuf, ent_res, N_ENTITIES);
    // user stage uses UNnormalized entity_agg (= summed)
    k_coef_scatter<<<NNZ / 8, 256, 0, stream>>>(iu, ii, vals, score, denom, summed, umean, 0);
    k_score<<<NNZ / 8, 256, 0, stream>>>(iu, ii, summed, umean, score, (unsigned*)rowmax);
    k_exp_denom<<<(NNZ + 255) / 256, 256, 0, stream>>>(iu, score, (const unsigned*)rowmax, denom);
    k_coef_scatter<<<NNZ / 8, 256, 0, stream>>>(iu, ii, vals, score, denom, summed, uagg, 1);
    k_norm_acc<<<(N_USERS / 16 + 7) / 8, 256, 0, stream>>>(uagg, nullptr, user_res, N_USERS);
    ent_in = entbuf;                                // next hop input = normalized ent
  }
}